// BertCrfForNer_35570919145974
// MI455X (gfx1250) — compile-verified
//
#include <hip/hip_runtime.h>
#include <hip/hip_bf16.h>
#include <math.h>

typedef __attribute__((ext_vector_type(2))) float v2f;
typedef __attribute__((ext_vector_type(8))) float v8f;

#define Bn 4
#define Pn 160
#define Ln 128
#define Hn 768
#define Tn 9

// ---------------------------------------------------------------- init
__global__ void init_ws_kernel(unsigned* minbits) {
    if (threadIdx.x == 0) *minbits = 0xFFFFFFFFu;  // max key == +inf float
}

// ------------------------------------------------- global min of bert_embs
// monotonic encoding: key(f) strictly increasing with f over all finite floats
__device__ __forceinline__ unsigned float_to_key(float f) {
    unsigned bits = __float_as_uint(f);
    return (bits & 0x80000000u) ? ~bits : (bits | 0x80000000u);
}
__device__ __forceinline__ float key_to_float(unsigned k) {
    unsigned bits = (k & 0x80000000u) ? (k ^ 0x80000000u) : ~k;
    return __uint_as_float(bits);
}

__global__ __launch_bounds__(256) void min_kernel(const float* __restrict__ bert,
                                                  int n, unsigned* minbits) {
    int idx = blockIdx.x * blockDim.x + threadIdx.x;
    int stride = gridDim.x * blockDim.x;
    float v = 3.402823466e38f;
    for (int i = idx; i < n; i += stride) v = fminf(v, bert[i]);
    // wave32 reduction
    for (int off = 16; off > 0; off >>= 1)
        v = fminf(v, __shfl_down(v, off, 32));
    __shared__ float s[8];
    int wave = threadIdx.x >> 5;
    if ((threadIdx.x & 31) == 0) s[wave] = v;
    __syncthreads();
    if (threadIdx.x == 0) {
        float m = s[0];
        for (int w = 1; w < (int)(blockDim.x >> 5); ++w) m = fminf(m, s[w]);
        atomicMin(minbits, float_to_key(m));
    }
}

// ---------------------------------------- pieces->word masked max pooling
// one block per word (b,l); 256 threads cover H=768 (3 each)
__global__ __launch_bounds__(256) void pool_kernel(const float* __restrict__ bert,
                                                   const unsigned char* __restrict__ p2w,
                                                   const unsigned char* __restrict__ masks,
                                                   const unsigned* __restrict__ minbits,
                                                   float* __restrict__ wr) {
    __shared__ unsigned char flags[Pn];
    __shared__ float s_min;
    const int m = blockIdx.x;          // word index in [0, B*L)
    const int b = m >> 7;              // /128
    const int l = m & 127;
    const int t = threadIdx.x;
    if (t < Pn) flags[t] = p2w[((size_t)(b * Ln + l)) * Pn + t];
    if (t == 0) s_min = key_to_float(*minbits);
    __syncthreads();

    const float mk = masks[b * Ln + l] ? 1.0f : 0.0f;
    const float mv = s_min;
    float a0 = mv, a1 = mv, a2 = mv;
    const float* base = bert + (size_t)b * Pn * Hn;
    for (int p = 0; p < Pn; ++p) {
        if (flags[p]) {                 // uniform branch across the block
            const float* row = base + (size_t)p * Hn;
            a0 = fmaxf(a0, row[t]);
            a1 = fmaxf(a1, row[t + 256]);
            a2 = fmaxf(a2, row[t + 512]);
        }
    }
    float* o = wr + (size_t)m * Hn;
    o[t]       = a0 * mk;
    o[t + 256] = a1 * mk;
    o[t + 512] = a2 * mk;
}

// ------------------------------------------------------- logits via WMMA f32
// logits[m][n] = sum_k wr[m][k] * W[k][n] + bias[n], M=512, N=9 (padded to 16), K=768
// One wave per 16-row tile; V_WMMA_F32_16X16X4_F32, 192 K-steps.
__global__ __launch_bounds__(32) void gemm_wmma_kernel(const float* __restrict__ wr,
                                                       const float* __restrict__ W,
                                                       const float* __restrict__ bias,
                                                       float* __restrict__ out) {
    const int lane = threadIdx.x;       // 0..31
    const int half = lane >> 4;         // 0 or 1
    const int lid  = lane & 15;
    const int mtile = blockIdx.x;       // 0..31
    const int m = mtile * 16 + lid;     // A row (same for both half-waves)

    // pad N: clamp address + zero-scale keeps loads full-EXEC (no divergence)
    const int   cn = (lid < Tn) ? lid : 0;
    const float sc = (lid < Tn) ? 1.0f : 0.0f;

    // C init: bias depends only on N=lid; all 8 accumulator rows share it
    const float bv = bias[cn] * sc;
    v8f c;
#pragma unroll
    for (int r = 0; r < 8; ++r) c[r] = bv;

    const float* arow = wr + (size_t)m * Hn;
    for (int k0 = 0; k0 < Hn; k0 += 4) {
        const int k = k0 + 2 * half;    // lanes 0-15: K={k0,k0+1}; 16-31: K={k0+2,k0+3}
        v2f a;
        a.x = arow[k];
        a.y = arow[k + 1];
        v2f bb;
        bb.x = W[(size_t)k * Tn + cn] * sc;
        bb.y = W[(size_t)(k + 1) * Tn + cn] * sc;
        c = __builtin_amdgcn_wmma_f32_16x16x4_f32(
            /*neg_a=*/false, a, /*neg_b=*/false, bb,
            /*c_mod=*/(short)0, c, /*reuse_a=*/false, /*reuse_b=*/false);
    }

    // D layout: VGPR r holds (M = r + 8*half, N = lid)
    if (lid < Tn) {
#pragma unroll
        for (int r = 0; r < 8; ++r) {
            const int mm = mtile * 16 + r + 8 * half;
            out[(size_t)mm * Tn + lid] = c[r];
        }
    }
}

// ----------------------------------------------------------------- CRF loss
// lane b handles batch b serially (the scan is a pure latency chain, T=9 tiny)
__global__ __launch_bounds__(32) void crf_kernel(const float* __restrict__ logits,
                                                 const int* __restrict__ labels,
                                                 const unsigned char* __restrict__ masks,
                                                 const float* __restrict__ startT,
                                                 const float* __restrict__ endT,
                                                 const float* __restrict__ trans,
                                                 float* __restrict__ loss_out) {
    __shared__ float part[Bn];
    const int b = threadIdx.x;
    if (b < Bn) {
        const float* em = logits + (size_t)b * Ln * Tn;
        const int* tg = labels + b * Ln;
        const unsigned char* mk = masks + b * Ln;

        float tr[Tn * Tn];
        for (int i = 0; i < Tn * Tn; ++i) tr[i] = trans[i];

        // numerator
        float num = startT[tg[0]] + em[tg[0]];
        // forward alphas
        float alpha[Tn];
        for (int t = 0; t < Tn; ++t) alpha[t] = startT[t] + em[t];

        int last = 0;
        for (int l = 1; l < Ln; ++l) {
            const float mf = mk[l] ? 1.0f : 0.0f;
            const int tp = tg[l - 1], tc = tg[l];
            num += (tr[tp * Tn + tc] + em[l * Tn + tc]) * mf;
            if (mk[l]) {
                last = l;
                float na[Tn];
                for (int j = 0; j < Tn; ++j) {
                    float mx = -3.402823466e38f;
                    for (int i = 0; i < Tn; ++i)
                        mx = fmaxf(mx, alpha[i] + tr[i * Tn + j]);
                    float s = 0.0f;
                    for (int i = 0; i < Tn; ++i)
                        s += expf(alpha[i] + tr[i * Tn + j] - mx);
                    na[j] = mx + logf(s) + em[l * Tn + j];
                }
                for (int j = 0; j < Tn; ++j) alpha[j] = na[j];
            }
        }
        num += endT[tg[last]];

        float mx = -3.402823466e38f;
        for (int j = 0; j < Tn; ++j) mx = fmaxf(mx, alpha[j] + endT[j]);
        float s = 0.0f;
        for (int j = 0; j < Tn; ++j) s += expf(alpha[j] + endT[j] - mx);
        const float den = mx + logf(s);

        part[b] = den - num;   // -llh
    }
    __syncthreads();
    if (threadIdx.x == 0)
        loss_out[0] = part[0] + part[1] + part[2] + part[3];
}

// ---------------------------------------------------------------- launcher
extern "C" void kernel_launch(void* const* d_in, const int* in_sizes, int n_in,
                              void* d_out, int out_size, void* d_ws, size_t ws_size,
                              hipStream_t stream) {
    const float*         bert   = (const float*)d_in[0];         // (B,P,H) f32
    const unsigned char* p2w    = (const unsigned char*)d_in[1]; // (B,L,P) bool
    const unsigned char* masks  = (const unsigned char*)d_in[2]; // (B,L)   bool
    const int*           labels = (const int*)d_in[3];           // (B,L)   i32
    /* d_in[4] = sent_length (unused; masks encode it) */
    const float*         W      = (const float*)d_in[5];         // (H,T) f32
    const float*         bias   = (const float*)d_in[6];         // (T,)
    const float*         startT = (const float*)d_in[7];         // (T,)
    const float*         endT   = (const float*)d_in[8];         // (T,)
    const float*         trans  = (const float*)d_in[9];         // (T,T)

    float* out = (float*)d_out;          // [0]=loss, [1..]=logits (B,L,T)

    unsigned* minbits = (unsigned*)d_ws;
    float*    wr      = (float*)((char*)d_ws + 256);   // word_reps [B*L][H]

    init_ws_kernel<<<1, 32, 0, stream>>>(minbits);
    min_kernel<<<256, 256, 0, stream>>>(bert, Bn * Pn * Hn, minbits);
    pool_kernel<<<Bn * Ln, 256, 0, stream>>>(bert, p2w, masks, minbits, wr);
    gemm_wmma_kernel<<<(Bn * Ln) / 16, 32, 0, stream>>>(wr, W, bias, out + 1);
    crf_kernel<<<1, 32, 0, stream>>>(out + 1, labels, masks, startT, endT, trans, out);
}